// Proximity_41197326303939
// MI455X (gfx1250) — compile-verified
//
#include <hip/hip_runtime.h>
#include <math.h>

typedef __attribute__((ext_vector_type(2))) float v2f;
typedef __attribute__((ext_vector_type(8))) float v8f;

#define WAVES_PER_BLOCK 8
#define BLOCK_THREADS   256
#define GRID_BLOCKS     512
#define ROW_PAD         68        // 64 floats + 4 pad -> conflict-free ds_load_b64
#define EPS_LO          1e-12f
#define EPS_HI          1e12f

// Each wave processes tiles of 16 samples (D = 64).
// diff tile staged in LDS; Gram matrix via V_WMMA_F32_16X16X4_F32 over 16 K-steps;
// diagonal = per-sample squared distance.
__global__ __launch_bounds__(BLOCK_THREADS)
void prox_main_kernel(const float* __restrict__ x,
                      const int*   __restrict__ labels,
                      const float* __restrict__ centers,
                      float*       __restrict__ partial,
                      int nSamples)
{
    __shared__ __align__(16) float ldsD[WAVES_PER_BLOCK][16 * ROW_PAD];
    __shared__ int   ldsLab[WAVES_PER_BLOCK][16];
    __shared__ float ldsRed[WAVES_PER_BLOCK];

    const int tid  = threadIdx.x;
    const int wave = tid >> 5;
    const int lane = tid & 31;

    const int nTiles       = nSamples >> 4;                 // 16 samples per tile
    const int tilesPerStep = GRID_BLOCKS * WAVES_PER_BLOCK; // wave-parallel tiles
    const int iters        = (nTiles + tilesPerStep - 1) / tilesPerStep;

    float acc = 0.0f;

    for (int it = 0; it < iters; ++it) {
        const int  tile   = it * tilesPerStep + (int)blockIdx.x * WAVES_PER_BLOCK + wave;
        const bool active = (tile < nTiles);
        const int  base   = tile << 4;

        // ---- phase 0: stage labels for this tile ----
        if (active && lane < 16) {
            ldsLab[wave][lane] = labels[base + lane];
        }
        __syncthreads();

        // ---- phase 1: coalesced b128 loads of x + gathered centers; store diff ----
        if (active) {
            #pragma unroll
            for (int i = 0; i < 8; ++i) {
                const int f   = i * 32 + lane;   // 0..255 float4 slots of the tile
                const int row = f >> 4;          // 0..15 sample within tile
                const int c4  = f & 15;          // 0..15 float4 within row
                const float4 xv =
                    ((const float4*)(x + (size_t)(base + row) * 64))[c4];
                const int crow = ldsLab[wave][row];
                const float4 cv =
                    ((const float4*)(centers + (size_t)crow * 64))[c4];
                float4 dv;
                dv.x = xv.x - cv.x; dv.y = xv.y - cv.y;
                dv.z = xv.z - cv.z; dv.w = xv.w - cv.w;
                *(float4*)&ldsD[wave][row * ROW_PAD + c4 * 4] = dv;
            }
        }
        __syncthreads();

        // ---- phase 2: Gram via WMMA f32 16x16x4; A and B share registers ----
        if (active) {
            v8f cacc = {};
            const int s   = lane & 15;         // sample within tile
            const int off = (lane >> 4) << 1;  // 0 (lanes 0-15) or 2 (lanes 16-31)
            #pragma unroll
            for (int t = 0; t < 16; ++t) {
                const int dim = 4 * t + off;
                const v2f ab  = *(const v2f*)&ldsD[wave][s * ROW_PAD + dim];
                // (neg_a, A, neg_b, B, c_mod, C, reuse_a, reuse_b)
                cacc = __builtin_amdgcn_wmma_f32_16x16x4_f32(
                    false, ab, false, ab, (short)0, cacc, false, false);
            }
            // diagonal: i<8 -> lane i, VGPR i;  i>=8 -> lane i+16 (24..31), VGPR lane-24
            float dsq = 0.0f;
            #pragma unroll
            for (int r = 0; r < 8; ++r) {
                if (lane == r || lane == r + 24) dsq = cacc[r];
            }
            if (lane < 8 || lane >= 24) {
                float d = sqrtf(dsq);
                d = fminf(fmaxf(d, EPS_LO), EPS_HI);
                acc += d;
            }
        }
        __syncthreads();
    }

    // ---- wave32 reduction, then block reduction ----
    for (int o = 16; o > 0; o >>= 1)
        acc += __shfl_xor(acc, o, 32);
    if (lane == 0) ldsRed[wave] = acc;
    __syncthreads();
    if (tid == 0) {
        float s = 0.0f;
        #pragma unroll
        for (int w = 0; w < WAVES_PER_BLOCK; ++w) s += ldsRed[w];
        partial[blockIdx.x] = s;
    }
}

// Deterministic final reduction: 512 partials -> mean
__global__ __launch_bounds__(256)
void prox_reduce_kernel(const float* __restrict__ partial,
                        float* __restrict__ out,
                        int n, float invB)
{
    __shared__ float red[256];
    const int tid = threadIdx.x;
    float v = 0.0f;
    for (int i = tid; i < n; i += 256) v += partial[i];
    red[tid] = v;
    __syncthreads();
    for (int s = 128; s > 0; s >>= 1) {
        if (tid < s) red[tid] += red[tid + s];
        __syncthreads();
    }
    if (tid == 0) out[0] = red[0] * invB;
}

extern "C" void kernel_launch(void* const* d_in, const int* in_sizes, int n_in,
                              void* d_out, int out_size, void* d_ws, size_t ws_size,
                              hipStream_t stream)
{
    const float* x       = (const float*)d_in[0];   // [B, 64] f32
    const int*   labels  = (const int*)  d_in[1];   // [B] int32
    const float* centers = (const float*)d_in[2];   // [1000, 64] f32
    float*       out     = (float*)d_out;           // scalar mean
    float*       partial = (float*)d_ws;            // GRID_BLOCKS floats scratch

    const int B = in_sizes[1];

    prox_main_kernel<<<GRID_BLOCKS, BLOCK_THREADS, 0, stream>>>(
        x, labels, centers, partial, B);
    prox_reduce_kernel<<<1, 256, 0, stream>>>(
        partial, out, GRID_BLOCKS, 1.0f / (float)B);
}